// GraphConvNet_16381005267349
// MI455X (gfx1250) — compile-verified
//
#include <hip/hip_runtime.h>
#include <math.h>

#define E     2048
#define S     1024
#define NHEAD 8
#define DH    256
#define NREL  8
#define DMAP  4096

typedef __attribute__((ext_vector_type(2))) float v2f;
typedef __attribute__((ext_vector_type(8))) float v8f;

// ---------------------------------------------------------------------------
// Conv2d(1,8,2) on 17x17 -> (8,16,16) flattened to h[2048]
// ---------------------------------------------------------------------------
__global__ __launch_bounds__(256) void conv_kernel(
    const float* __restrict__ x, const float* __restrict__ w,
    const float* __restrict__ b, float* __restrict__ h)
{
    int e = blockIdx.x * 256 + threadIdx.x;   // 8 blocks x 256 = 2048
    int c  = e >> 8;
    int r  = (e >> 4) & 15;
    int cl = e & 15;
    float acc = b[c];
    acc += x[(r + 0) * 17 + cl + 0] * w[c * 4 + 0];
    acc += x[(r + 0) * 17 + cl + 1] * w[c * 4 + 1];
    acc += x[(r + 1) * 17 + cl + 0] * w[c * 4 + 2];
    acc += x[(r + 1) * 17 + cl + 1] * w[c * 4 + 3];
    h[e] = acc;
}

// ---------------------------------------------------------------------------
// Generic matvec: y[row] = (relu?) dot(W[row,:], x) + b[row]
// one wave (32 lanes) per row, 8 rows per 256-thread block, float4 streaming
// ---------------------------------------------------------------------------
__global__ __launch_bounds__(256) void matvec_kernel(
    const float* __restrict__ W, const float* __restrict__ x,
    const float* __restrict__ b, float* __restrict__ y,
    int rows, int do_relu)
{
    __shared__ __align__(16) float xs[E];
    int tid = threadIdx.x;
    for (int i = tid; i < E; i += 256) xs[i] = x[i];
    __syncthreads();

    int wave = tid >> 5;
    int lane = tid & 31;
    int row  = blockIdx.x * 8 + wave;
    if (row < rows) {
        const float4* wr4 = (const float4*)(W + (size_t)row * E);
        const float4* xs4 = (const float4*)xs;
        float acc = 0.f;
        for (int c = lane; c < E / 4; c += 32) {
            float4 a = wr4[c];
            float4 v = xs4[c];
            acc += a.x * v.x + a.y * v.y + a.z * v.z + a.w * v.w;
        }
        for (int off = 16; off > 0; off >>= 1)
            acc += __shfl_xor(acc, off, 32);
        if (lane == 0) {
            float r = acc + b[row];
            if (do_relu) r = fmaxf(r, 0.f);
            y[row] = r;
        }
    }
}

// ---------------------------------------------------------------------------
// K/V projection GEMM via V_WMMA_F32_16X16X4_F32 (fp32, reference precision).
//   K = neighbors @ wk^T + bk ; V = neighbors @ wv^T + bv
// D[m,n] = sum_k A[m,k] * W[n,k]   (B[k,n] = W[n,k] -> two consecutive floats
// of row n per lane, one b64 global load per WMMA step).
// grid = (64 s-tiles, 32 job groups), 8 waves/block, 1 job (16x16 tile)/wave.
// ---------------------------------------------------------------------------
__global__ __launch_bounds__(256) void gemm_kv_kernel(
    const float* __restrict__ A,    // neighbors (S x E)
    const float* __restrict__ Wkv,  // [wk ; wv] (2E x E) row-major
    const float* __restrict__ Bkv,  // [bk ; bv] (2E)
    float* __restrict__ Kout, float* __restrict__ Vout)
{
    __shared__ __align__(16) float As[16 * 132];  // 16 x 128 chunk, pad->bank-safe

    const int tid  = threadIdx.x;
    const int wave = tid >> 5;
    const int lane = tid & 31;
    const int n    = lane & 15;       // N index (and A-matrix M index)
    const int ksel = lane >> 4;       // which K pair this half-wave owns

    const int s0  = blockIdx.x * 16;
    const int job = blockIdx.y * 8 + wave;   // 0..255
    const int mat = job >> 7;                // 0=K, 1=V
    const int e0  = (job & 127) * 16;

    const float* Wrow = Wkv + (size_t)mat * E * E + (size_t)(e0 + n) * E;
    float* Out = mat ? Vout : Kout;

    v8f acc = {};

    for (int kk0 = 0; kk0 < E; kk0 += 128) {
        // cooperative stage of A[s0..s0+15][kk0..kk0+127] into LDS
        {
            int f = tid * 8;
            int m = f >> 7;
            int c = f & 127;
            const float4* src = (const float4*)(A + (size_t)(s0 + m) * E + kk0 + c);
            float4 p0 = src[0];
            float4 p1 = src[1];
            *(float4*)&As[m * 132 + c + 0] = p0;
            *(float4*)&As[m * 132 + c + 4] = p1;
        }
        if (kk0 + 128 < E)  // speculative prefetch of next weight chunk
            __builtin_prefetch(Wrow + kk0 + 128 + 2 * ksel, 0, 1);
        __syncthreads();

        const float* arow = &As[n * 132];
        #pragma unroll 4
        for (int st = 0; st < 32; ++st) {
            v2f a = *(const v2f*)&arow[st * 4 + 2 * ksel];                 // ds_load_b64
            v2f b = *(const v2f*)&Wrow[kk0 + st * 4 + 2 * ksel];          // global b64
            acc = __builtin_amdgcn_wmma_f32_16x16x4_f32(
                false, a, false, b, (short)0, acc, false, false);
        }
        __syncthreads();
    }

    const float bias = Bkv[mat * E + e0 + n];
    size_t base = (size_t)(s0 + 8 * ksel) * E + e0 + n;  // row = s0 + p + 8*ksel
    #pragma unroll
    for (int p = 0; p < 8; ++p)
        Out[base + (size_t)p * E] = acc[p] + bias;
}

// ---------------------------------------------------------------------------
// Single-query MHA: one block per head. scores -> softmax -> weighted V sum.
// ---------------------------------------------------------------------------
__global__ __launch_bounds__(256) void attn_kernel(
    const float* __restrict__ q, const float* __restrict__ k,
    const float* __restrict__ v, float* __restrict__ o)
{
    __shared__ __align__(16) float qs[DH];
    __shared__ float sc[S];
    __shared__ float red[256];

    const int hh  = blockIdx.x;
    const int tid = threadIdx.x;
    qs[tid] = q[hh * DH + tid];
    __syncthreads();

    const float scale = 0.0625f;  // DH^-0.5 = 1/16
    for (int s = tid; s < S; s += 256) {
        const float4* kr = (const float4*)(k + (size_t)s * E + hh * DH);
        const float4* qq = (const float4*)qs;
        float acc = 0.f;
        for (int d = 0; d < DH / 4; ++d) {
            float4 a = kr[d];
            float4 b = qq[d];
            acc += a.x * b.x + a.y * b.y + a.z * b.z + a.w * b.w;
        }
        sc[s] = acc * scale;
    }
    __syncthreads();

    // max
    float m = -1e30f;
    for (int s = tid; s < S; s += 256) m = fmaxf(m, sc[s]);
    red[tid] = m;
    __syncthreads();
    for (int st = 128; st > 0; st >>= 1) {
        if (tid < st) red[tid] = fmaxf(red[tid], red[tid + st]);
        __syncthreads();
    }
    m = red[0];
    __syncthreads();

    // exp + sum
    float sum = 0.f;
    for (int s = tid; s < S; s += 256) {
        float e = expf(sc[s] - m);
        sc[s] = e;
        sum += e;
    }
    red[tid] = sum;
    __syncthreads();
    for (int st = 128; st > 0; st >>= 1) {
        if (tid < st) red[tid] += red[tid + st];
        __syncthreads();
    }
    const float inv = 1.f / red[0];
    __syncthreads();

    // o[d] = sum_s attn[s] * v[s, hh*DH + d]   (coalesced over d = tid)
    float acc = 0.f;
    const float* vc = v + hh * DH + tid;
    for (int s = 0; s < S; ++s)
        acc += sc[s] * vc[(size_t)s * E];
    o[hh * DH + tid] = acc * inv;
}

// ---------------------------------------------------------------------------
// Final softmax over 4096 logits (single block)
// ---------------------------------------------------------------------------
__global__ __launch_bounds__(256) void softmax4096_kernel(
    const float* __restrict__ logits, float* __restrict__ out)
{
    __shared__ float buf[DMAP];
    __shared__ float red[256];
    const int tid = threadIdx.x;

    float m = -1e30f;
    for (int i = tid; i < DMAP; i += 256) {
        float v = logits[i];
        buf[i] = v;
        m = fmaxf(m, v);
    }
    red[tid] = m;
    __syncthreads();
    for (int st = 128; st > 0; st >>= 1) {
        if (tid < st) red[tid] = fmaxf(red[tid], red[tid + st]);
        __syncthreads();
    }
    m = red[0];
    __syncthreads();

    float sum = 0.f;
    for (int i = tid; i < DMAP; i += 256) {
        float e = expf(buf[i] - m);
        buf[i] = e;
        sum += e;
    }
    red[tid] = sum;
    __syncthreads();
    for (int st = 128; st > 0; st >>= 1) {
        if (tid < st) red[tid] += red[tid + st];
        __syncthreads();
    }
    const float inv = 1.f / red[0];
    __syncthreads();
    for (int i = tid; i < DMAP; i += 256)
        out[i] = buf[i] * inv;
}

// ---------------------------------------------------------------------------
extern "C" void kernel_launch(void* const* d_in, const int* in_sizes, int n_in,
                              void* d_out, int out_size, void* d_ws, size_t ws_size,
                              hipStream_t stream)
{
    const float* x          = (const float*)d_in[0];
    const float* neighbors  = (const float*)d_in[1];
    const float* conv_w     = (const float*)d_in[2];
    const float* conv_b     = (const float*)d_in[3];
    const float* in_proj_w  = (const float*)d_in[4];
    const float* in_proj_b  = (const float*)d_in[5];
    const float* out_proj_w = (const float*)d_in[6];
    const float* out_proj_b = (const float*)d_in[7];
    const float* ff_w       = (const float*)d_in[8];
    const float* ff_b       = (const float*)d_in[9];
    const float* out_w      = (const float*)d_in[10];
    const float* out_b      = (const float*)d_in[11];

    float* ws     = (float*)d_ws;
    float* h      = ws;              // 2048
    float* q      = ws + 2048;       // 2048
    float* o      = ws + 4096;       // 2048
    float* h2     = ws + 6144;       // 2048
    float* logits = ws + 8192;       // 4096
    float* kbuf   = ws + 12288;      // S*E
    float* vbuf   = kbuf + (size_t)S * E;

    conv_kernel<<<8, 256, 0, stream>>>(x, conv_w, conv_b, h);

    for (int r = 0; r < NREL; ++r) {
        const float* wq  = in_proj_w + (size_t)r * 3 * E * E;
        const float* wkv = wq + (size_t)E * E;        // [wk ; wv]
        const float* bq  = in_proj_b + (size_t)r * 3 * E;

        matvec_kernel<<<E / 8, 256, 0, stream>>>(wq, h, bq, q, E, 0);
        gemm_kv_kernel<<<dim3(64, 32), 256, 0, stream>>>(neighbors, wkv, bq + E,
                                                         kbuf, vbuf);
        attn_kernel<<<NHEAD, 256, 0, stream>>>(q, kbuf, vbuf, o);
        matvec_kernel<<<E / 8, 256, 0, stream>>>(out_proj_w + (size_t)r * E * E, o,
                                                 out_proj_b + (size_t)r * E, h2, E, 0);
        matvec_kernel<<<E / 8, 256, 0, stream>>>(ff_w + (size_t)r * E * E, h2,
                                                 ff_b + (size_t)r * E, h, E, 1);
    }

    matvec_kernel<<<DMAP / 8, 256, 0, stream>>>(out_w, h, out_b, logits, DMAP, 0);
    softmax4096_kernel<<<1, 256, 0, stream>>>(logits, (float*)d_out);
}